// EnhancedQuantumAttention_45286135169502
// MI455X (gfx1250) — compile-verified
//
#include <hip/hip_runtime.h>
#include <math.h>

// ---------------------------------------------------------------------------
// EnhancedQuantumAttention for MI455X (gfx1250, wave32).
// Fused flash-attention-style kernel, fp32 WMMA (v_wmma_f32_16x16x4_f32).
// One wave = 16 query rows of one (b,h); 4 scale freqs looped in-register.
// No LDS, no barriers, no workspace: phase patterns / norms / expert
// modulation are computed analytically on the fly.
// ---------------------------------------------------------------------------

typedef float v2f __attribute__((ext_vector_type(2)));
typedef float v8f __attribute__((ext_vector_type(8)));

#define PI_F 3.14159265358979323846f

__device__ __forceinline__ float lane_get(float v, int srcLane) {
  // wave32 gather: dst[lane] = v[srcLane]
  return __int_as_float(__builtin_amdgcn_ds_bpermute(srcLane << 2, __float_as_int(v)));
}

__device__ __forceinline__ v8f wmma_k4(v2f a, v2f b, v8f c) {
  // D(16x16,f32) = A(16x4,f32) * B(4x16,f32) + C
  return __builtin_amdgcn_wmma_f32_16x16x4_f32(
      /*neg_a=*/false, a, /*neg_b=*/false, b,
      /*c_mod=*/(short)0, c, /*reuse_a=*/false, /*reuse_b=*/false);
}

// Dirichlet kernel: sum_{l=0}^{2047} cos(alpha*l)
__device__ __forceinline__ float dsum2048(float alpha) {
  return __cosf(1023.5f * alpha) * __sinf(1024.0f * alpha) / __sinf(0.5f * alpha);
}

__global__ __launch_bounds__(256)
void eqa_kernel(const float* __restrict__ Qr, const float* __restrict__ Qi,
                const float* __restrict__ Kr, const float* __restrict__ Ki,
                const float* __restrict__ Vr, const float* __restrict__ Vi,
                float* __restrict__ out) {
  constexpr int L = 1024, D = 64;
  constexpr size_t NOUT = (size_t)2 * 16 * L * D;   // one complex component plane

  const int lane = threadIdx.x & 31;
  const int wave = threadIdx.x >> 5;        // 8 waves
  const int hi   = (lane >> 4) & 1;         // lane half (0: lanes 0-15, 1: 16-31)
  const int col  = lane & 15;
  const int hi8  = hi << 3;

  const int qchunk = blockIdx.x & 7;        // 8 q-chunks of 128 rows
  const int bh     = blockIdx.x >> 3;       // 0..31
  const int h      = bh & 15;
  const int q0     = qchunk * 128 + wave * 16;

  const size_t base = (size_t)bh * L * D;
  const float inv_sqrt_d = 0.125f;

  // head phase
  float sph, cph;
  __sincosf(2.0f * PI_F * (float)h / 16.0f, &sph, &cph);

  // scale-pattern value at row l: pr + i*pi = e^{i*ph} * (e^{it}+e^{i2t}+e^{it/2}) / norm
  auto pattern = [&](int l, float w, float invn, float& pr, float& pi_) {
    float t = w * (float)l;
    float s1, c1, s2, c2, s3, c3;
    __sincosf(t,        &s1, &c1);
    __sincosf(2.0f * t, &s2, &c2);
    __sincosf(0.5f * t, &s3, &c3);
    float Sx = c1 + c2 + c3, Sy = s1 + s2 + s3;
    pr  = (cph * Sx - sph * Sy) * invn;
    pi_ = (sph * Sx + cph * Sy) * invn;
  };

  const v8f zero8 = {0.f, 0.f, 0.f, 0.f, 0.f, 0.f, 0.f, 0.f};
  v8f totR[4] = {zero8, zero8, zero8, zero8};
  v8f totI[4] = {zero8, zero8, zero8, zero8};

  const float freqs4[4] = {1.0f, 0.5f, 0.25f, 0.1f};

  for (int sc = 0; sc < 4; ++sc) {
    const float f = freqs4[sc];
    const float w = 2.0f * PI_F * f / 2047.0f;
    // norm^2 = sum_l (3 + 2[cos t + cos t/2 + cos 3t/2])  (head-independent)
    const float n2 = 3.0f * 2048.0f +
                     2.0f * (dsum2048(w) + dsum2048(0.5f * w) + dsum2048(1.5f * w));
    const float invn = 1.0f / sqrtf(n2);

    // ---- stationary modulated Q as B-fragments (B[d,l] = qp[q0+l, d]) ----
    float qpr, qpi;
    pattern(q0 + col, w, invn, qpr, qpi);
    const float* qrRow = Qr + base + (size_t)(q0 + col) * D + 2 * hi;
    const float* qiRow = Qi + base + (size_t)(q0 + col) * D + 2 * hi;
    v2f bqr[16], bqi[16];
#pragma unroll
    for (int kk = 0; kk < 16; ++kk) {
      v2f a = *(const v2f*)(qrRow + 4 * kk);
      v2f b = *(const v2f*)(qiRow + 4 * kk);
      bqr[kk] = a * qpr - b * qpi;
      bqi[kk] = a * qpi + b * qpr;
    }

    v8f accR[4] = {zero8, zero8, zero8, zero8};
    v8f accI[4] = {zero8, zero8, zero8, zero8};
    float m_run = -__builtin_inff();
    float d_run = 0.0f;

    for (int mb = 0; mb < 64; ++mb) {
      const int m0 = mb * 16;
      float kc, ks;
      pattern(m0 + col, w, invn, kc, ks);
      const float* krRow = Kr + base + (size_t)(m0 + col) * D + 2 * hi;
      const float* kiRow = Ki + base + (size_t)(m0 + col) * D + 2 * hi;

      // ---- scores (transposed): srT[m,l], siT[m,l] ----
      v8f srT = zero8, siT = zero8;
#pragma unroll
      for (int kk = 0; kk < 16; ++kk) {
        v2f a = *(const v2f*)(krRow + 4 * kk);
        v2f b = *(const v2f*)(kiRow + 4 * kk);
        v2f akr  = a * kc - b * ks;   // modulated K real  (A-fragment)
        v2f aki  = a * ks + b * kc;   // modulated K imag
        v2f akin = -aki;              // f32 WMMA has no A-neg modifier
        srT = wmma_k4(akr,  bqr[kk], srT);
        srT = wmma_k4(akin, bqi[kk], srT);
        siT = wmma_k4(akr,  bqi[kk], siT);
        siT = wmma_k4(aki,  bqr[kk], siT);
      }

      // ---- magnitude + online softmax over m (keys) ----
      v8f pv;
#pragma unroll
      for (int v = 0; v < 8; ++v)
        pv[v] = sqrtf(srT[v] * srT[v] + siT[v] * siT[v]) * inv_sqrt_d;

      float bmax = pv[0];
#pragma unroll
      for (int v = 1; v < 8; ++v) bmax = fmaxf(bmax, pv[v]);
      bmax = fmaxf(bmax, lane_get(bmax, lane ^ 16));   // other 8 m's for same query
      const float m_new = fmaxf(m_run, bmax);

      float bsum = 0.0f;
#pragma unroll
      for (int v = 0; v < 8; ++v) {
        pv[v] = __expf(pv[v] - m_new);
        bsum += pv[v];
      }
      bsum += lane_get(bsum, lane ^ 16);
      const float corr = __expf(m_run - m_new);        // exp(-inf)=0 on first block
      d_run = d_run * corr + bsum;
      m_run = m_new;

      // rescale accumulators: out-row of C element (vgpr v) is l' = v + 8*hi
      v8f fvec;
#pragma unroll
      for (int v = 0; v < 8; ++v) fvec[v] = lane_get(corr, v + hi8);
#pragma unroll
      for (int dd = 0; dd < 4; ++dd) { accR[dd] *= fvec; accI[dd] *= fvec; }

      // ---- P (C-layout, transposed) -> A-fragments of P(16l x 16m) ----
      v2f aP[4];
#pragma unroll
      for (int c = 0; c < 4; ++c) {
        const int src = col + ((c >= 2) ? 16 : 0);     // source half holds rows m
        float xa = lane_get(pv[(4 * c)     & 7], src); // for dest lanes hi=0
        float xb = lane_get(pv[(4 * c + 2) & 7], src); // for dest lanes hi=1
        float ya = lane_get(pv[(4 * c + 1) & 7], src);
        float yb = lane_get(pv[(4 * c + 3) & 7], src);
        aP[c].x = hi ? xb : xa;
        aP[c].y = hi ? yb : ya;
      }

      // ---- acc += P * V (V unmodulated) ----
      const float* vrBase = Vr + base + (size_t)m0 * D;
      const float* viBase = Vi + base + (size_t)m0 * D;
#pragma unroll
      for (int dd = 0; dd < 4; ++dd) {
#pragma unroll
        for (int c = 0; c < 4; ++c) {
          const int r = 4 * c + 2 * hi;
          const float* pr_ = vrBase + (size_t)r * D + 16 * dd + col;
          const float* pi2 = viBase + (size_t)r * D + 16 * dd + col;
          v2f bvr = {pr_[0], pr_[D]};
          v2f bvi = {pi2[0], pi2[D]};
          accR[dd] = wmma_k4(aP[c], bvr, accR[dd]);
          accI[dd] = wmma_k4(aP[c], bvi, accI[dd]);
        }
      }
    } // mb

    // ---- normalize by softmax denom and accumulate across scales ----
    const float inv = 1.0f / d_run;
    v8f ivec;
#pragma unroll
    for (int v = 0; v < 8; ++v) ivec[v] = lane_get(inv, v + hi8);
#pragma unroll
    for (int dd = 0; dd < 4; ++dd) {
      totR[dd] += accR[dd] * ivec;
      totI[dd] += accI[dd] * ivec;
    }
  } // scale

  // ---- epilogue: expert modulation + store ----
  // Er(l,d) = (cos(phd)*C_l - sin(phd)*S_l)/den,  Ei = (sin(phd)*C_l + cos(phd)*S_l)/den
  const float eden = 1.0f / (sqrtf(2048.0f) * sqrtf(24.0f));
  float cb[4], sb[4];
#pragma unroll
  for (int dd = 0; dd < 4; ++dd) {
    const int d = 16 * dd + col;
    __sincosf(2.0f * PI_F * (float)d / 64.0f, &sb[dd], &cb[dd]);
  }

#pragma unroll
  for (int v = 0; v < 8; ++v) {
    const int l = q0 + v + hi8;
    const float t = 2.0f * PI_F * (float)l / 2047.0f;
    float Cl = 0.0f, Sl = 0.0f;
#pragma unroll
    for (int i = 0; i < 8; ++i) {
      const float b0 = 0.3f + 0.1f * (float)i;
      const float b1 = 0.2f + 0.1f * (float)i;
      const float b2 = 0.1f + 0.1f * (float)i;
      float s, c;
      __sincosf(b0 * t, &s, &c); Cl += c; Sl += s;
      __sincosf(b1 * t, &s, &c); Cl += c; Sl += s;
      __sincosf(b2 * t, &s, &c); Cl += c; Sl += s;
    }
#pragma unroll
    for (int dd = 0; dd < 4; ++dd) {
      const float Er = (cb[dd] * Cl - sb[dd] * Sl) * eden;
      const float Ei = (sb[dd] * Cl + cb[dd] * Sl) * eden;
      const float ar = 0.5f * totR[dd][v];
      const float ai = 0.5f * totI[dd][v];
      const size_t idx = base + (size_t)l * D + 16 * dd + col;
      out[idx]        = ar * Er - ai * Ei;   // out_r plane
      out[NOUT + idx] = ar * Ei + ai * Er;   // out_i plane
    }
  }
}

extern "C" void kernel_launch(void* const* d_in, const int* in_sizes, int n_in,
                              void* d_out, int out_size, void* d_ws, size_t ws_size,
                              hipStream_t stream) {
  (void)in_sizes; (void)n_in; (void)out_size; (void)d_ws; (void)ws_size;
  const float* Qr = (const float*)d_in[0];
  const float* Qi = (const float*)d_in[1];
  const float* Kr = (const float*)d_in[2];
  const float* Ki = (const float*)d_in[3];
  const float* Vr = (const float*)d_in[4];
  const float* Vi = (const float*)d_in[5];
  // 32 (b,h) pairs * 8 q-chunks of 128 rows = 256 workgroups of 8 waves
  eqa_kernel<<<dim3(256), dim3(256), 0, stream>>>(Qr, Qi, Kr, Ki, Vr, Vi, (float*)d_out);
}